// RelationLayer_10866267259537
// MI455X (gfx1250) — compile-verified
//
#include <hip/hip_runtime.h>
#include <math.h>

// GAT-style relation layer for MI455X (gfx1250, wave32).
// FLOPs are trivial; the working set (~30 MB) fits in 192 MB L2, so the design
// goal is: exact-fp32 WMMA for the projection, rank-1 decomposition of the
// attention dot to kill per-edge feature gathers, and fully-parallel L2 atomics
// for the segment softmax + scatter.

typedef float v2f __attribute__((ext_vector_type(2)));
typedef float v8f __attribute__((ext_vector_type(8)));

#define N_DIM   128
#define OUT_DIM 64   // HEAD*HD = 4*16
#define HEADS   4
#define HD      16

// ---------------------------------------------------------------------------
// Kernel 1: hw = h @ W_w + b_w using V_WMMA_F32_16X16X4_F32.
// One wave computes one 16x16 output tile; 4 waves/block cover the 4 column
// tiles of the 64-wide output for a common 16-row slab of nodes.
// A layout (16x4 f32): lane L -> row m=L&15, K-pair base kb=2*(L>>4); a={K=kb,K=kb+1}
// B layout (4x16 f32): lane L -> col n=L&15, b={K=kb row, K=kb+1 row}
// C/D layout (16x16 f32): VGPR r, lanes0-15 -> M=r, lanes16-31 -> M=r+8.
// ---------------------------------------------------------------------------
__global__ void __launch_bounds__(128)
proj_wmma_kernel(const float* __restrict__ H, const float* __restrict__ Ww,
                 const float* __restrict__ bw, float* __restrict__ hw, int nNodes)
{
    const int lane = threadIdx.x & 31;
    const int wave = threadIdx.x >> 5;       // 0..3 -> output column tile
    const int row0 = blockIdx.x * 16;
    const int col0 = wave * 16;
    const int m    = lane & 15;
    const int n    = lane & 15;
    const int kb   = (lane >> 4) << 1;       // 0 or 2

    const int rowA = (row0 + m < nNodes) ? (row0 + m) : (nNodes - 1);
    const float* __restrict__ Arow = H + (size_t)rowA * N_DIM;

    v8f acc = {0.f, 0.f, 0.f, 0.f, 0.f, 0.f, 0.f, 0.f};

    #pragma unroll 8
    for (int k0 = 0; k0 < N_DIM; k0 += 4) {
        v2f a, b;
        a.x = Arow[k0 + kb];
        a.y = Arow[k0 + kb + 1];
        b.x = Ww[(size_t)(k0 + kb)     * OUT_DIM + col0 + n];
        b.y = Ww[(size_t)(k0 + kb + 1) * OUT_DIM + col0 + n];
        // 8 args: (neg_a, A, neg_b, B, c_mod, C, reuse_a, reuse_b)
        acc = __builtin_amdgcn_wmma_f32_16x16x4_f32(
                  false, a, false, b, (short)0, acc, false, false);
    }

    const float bias = bw[col0 + n];
    // Lane's store base: row = row0 + 8*(lane>=16) + r, col = col0 + n.
    const int   mrow0 = ((lane >> 4) << 3);
    float* __restrict__ base =
        hw + (size_t)(row0 + mrow0) * OUT_DIM + col0 + n;

    if (row0 + 16 <= nNodes) {
        // Whole tile in range (always true when nNodes % 16 == 0):
        // 8 unguarded stores at immediate offsets r*OUT_DIM.
        #pragma unroll
        for (int r = 0; r < 8; ++r)
            base[(size_t)r * OUT_DIM] = acc[r] + bias;
    } else {
        #pragma unroll
        for (int r = 0; r < 8; ++r)
            if (row0 + mrow0 + r < nNodes)
                base[(size_t)r * OUT_DIM] = acc[r] + bias;
    }
}

// ---------------------------------------------------------------------------
// Kernel 2: per (node, head) rank-1 attention projections + state init.
//   a_src[n,h] = hw[n,h,:] . W_a[0:16]
//   a_dst[n,h] = hw[n,h,:] . W_a[16:32]
// Also: m = -inf, denom = 0, out = 0.
// ---------------------------------------------------------------------------
__global__ void __launch_bounds__(256)
node_prep_kernel(const float* __restrict__ hw, const float* __restrict__ Wa,
                 float* __restrict__ a_src, float* __restrict__ a_dst,
                 float* __restrict__ mmax, float* __restrict__ denom,
                 float* __restrict__ out, int nNodes)
{
    const int t = blockIdx.x * blockDim.x + threadIdx.x;
    if (t >= nNodes * HEADS) return;
    const int n = t >> 2;
    const int h = t & 3;

    const float* __restrict__ v = hw + (size_t)n * OUT_DIM + h * HD;
    float s0 = 0.f, s1 = 0.f;
    #pragma unroll
    for (int d = 0; d < HD; ++d) {
        const float x = v[d];
        s0 = fmaf(x, Wa[d],      s0);
        s1 = fmaf(x, Wa[HD + d], s1);
    }
    a_src[t] = s0;
    a_dst[t] = s1;
    mmax[t]  = -INFINITY;
    denom[t] = 0.f;

    float* __restrict__ o = out + (size_t)n * OUT_DIM + h * HD;
    #pragma unroll
    for (int d = 0; d < HD; ++d) o[d] = 0.f;
}

// Lock-free float atomic-max via sign-split int/uint monotone mapping.
__device__ __forceinline__ void atomicMaxF(float* addr, float v)
{
    if (v >= 0.f) atomicMax((int*)addr, __float_as_int(v));
    else          atomicMin((unsigned int*)addr, __float_as_uint(v));
}

// ---------------------------------------------------------------------------
// Kernel 3: per-edge logits (all 4 heads) + segment max via atomics.
// ---------------------------------------------------------------------------
__global__ void __launch_bounds__(256)
edge_alpha_kernel(const int* __restrict__ src, const int* __restrict__ dst,
                  const float* __restrict__ a_src, const float* __restrict__ a_dst,
                  const float* __restrict__ ba,
                  float* __restrict__ alpha, float* __restrict__ mmax, int nE)
{
    const int e = blockIdx.x * blockDim.x + threadIdx.x;
    if (e >= nE) return;
    const int s = src[e];
    const int d = dst[e];
    const float4 as = *reinterpret_cast<const float4*>(a_src + (size_t)s * HEADS);
    const float4 ad = *reinterpret_cast<const float4*>(a_dst + (size_t)d * HEADS);
    const float  b  = ba[0];

    float4 al;
    al.x = as.x + ad.x + b;  al.x = (al.x > 0.f) ? al.x : 0.01f * al.x;
    al.y = as.y + ad.y + b;  al.y = (al.y > 0.f) ? al.y : 0.01f * al.y;
    al.z = as.z + ad.z + b;  al.z = (al.z > 0.f) ? al.z : 0.01f * al.z;
    al.w = as.w + ad.w + b;  al.w = (al.w > 0.f) ? al.w : 0.01f * al.w;

    *reinterpret_cast<float4*>(alpha + (size_t)e * HEADS) = al;

    float* mrow = mmax + (size_t)d * HEADS;
    atomicMaxF(mrow + 0, al.x);
    atomicMaxF(mrow + 1, al.y);
    atomicMaxF(mrow + 2, al.z);
    atomicMaxF(mrow + 3, al.w);
}

// ---------------------------------------------------------------------------
// Kernel 4: e = exp(alpha - m[dst]); segment sum of e via atomics.
// Overwrites alpha buffer with e in place.
// ---------------------------------------------------------------------------
__global__ void __launch_bounds__(256)
edge_exp_kernel(const int* __restrict__ dst, float* __restrict__ alpha,
                const float* __restrict__ mmax, float* __restrict__ denom, int nE)
{
    const int e = blockIdx.x * blockDim.x + threadIdx.x;
    if (e >= nE) return;
    const int d = dst[e];
    float4 al = *reinterpret_cast<float4*>(alpha + (size_t)e * HEADS);
    const float4 mm = *reinterpret_cast<const float4*>(mmax + (size_t)d * HEADS);

    al.x = __expf(al.x - mm.x);
    al.y = __expf(al.y - mm.y);
    al.z = __expf(al.z - mm.z);
    al.w = __expf(al.w - mm.w);

    *reinterpret_cast<float4*>(alpha + (size_t)e * HEADS) = al;

    float* drow = denom + (size_t)d * HEADS;
    atomicAdd(drow + 0, al.x);
    atomicAdd(drow + 1, al.y);
    atomicAdd(drow + 2, al.z);
    atomicAdd(drow + 3, al.w);
}

// ---------------------------------------------------------------------------
// Kernel 5: out[dst, j] += (e/denom) * hw[src, j].  One thread per (edge, j):
// 64 consecutive threads share an edge -> broadcast index/weight loads,
// contiguous hw loads, one f32 L2 atomic per thread (51.2M total, fully
// parallel so atomic latency is hidden by occupancy).
// ---------------------------------------------------------------------------
__global__ void __launch_bounds__(256)
edge_scatter_kernel(const int* __restrict__ src, const int* __restrict__ dst,
                    const float* __restrict__ ebuf, const float* __restrict__ denom,
                    const float* __restrict__ hw, float* __restrict__ out, int nE)
{
    const int t = blockIdx.x * blockDim.x + threadIdx.x;
    if (t >= nE * OUT_DIM) return;
    const int e = t >> 6;        // /64
    const int j = t & 63;
    const int h = j >> 4;        // head
    const int s = src[e];
    const int d = dst[e];

    const float w = ebuf[(size_t)e * HEADS + h] / denom[(size_t)d * HEADS + h];
    atomicAdd(out + (size_t)d * OUT_DIM + j, w * hw[(size_t)s * OUT_DIM + j]);
}

// ---------------------------------------------------------------------------
// Host launcher.  Inputs (setup_inputs order):
//   0:h [N,128] f32  1:W_w [128,64] f32  2:b_w [64] f32
//   3:W_a [32,1] f32 4:b_a [1] f32       5:src [E] i32   6:dst [E] i32
// Output: [N, 64] f32.
// ---------------------------------------------------------------------------
extern "C" void kernel_launch(void* const* d_in, const int* in_sizes, int n_in,
                              void* d_out, int out_size, void* d_ws, size_t ws_size,
                              hipStream_t stream)
{
    const float* h   = (const float*)d_in[0];
    const float* Ww  = (const float*)d_in[1];
    const float* bw  = (const float*)d_in[2];
    const float* Wa  = (const float*)d_in[3];
    const float* ba  = (const float*)d_in[4];
    const int*   src = (const int*)d_in[5];
    const int*   dst = (const int*)d_in[6];
    float*       out = (float*)d_out;

    const int nNodes = in_sizes[0] / N_DIM;   // 50000
    const int nE     = in_sizes[5];           // 800000

    // Workspace layout (floats), all 16B-aligned:
    float* ws    = (float*)d_ws;
    float* hw    = ws;                                  // N*64
    float* asrc  = hw    + (size_t)nNodes * OUT_DIM;    // N*4
    float* adst  = asrc  + (size_t)nNodes * HEADS;      // N*4
    float* mmax  = adst  + (size_t)nNodes * HEADS;      // N*4
    float* denom = mmax  + (size_t)nNodes * HEADS;      // N*4
    float* alpha = denom + (size_t)nNodes * HEADS;      // E*4 (then reused as e)

    // 1) Projection GEMM with fp32 WMMA. 50000/16 = 3125 row slabs.
    const int rowTiles = (nNodes + 15) / 16;
    proj_wmma_kernel<<<rowTiles, 128, 0, stream>>>(h, Ww, bw, hw, nNodes);

    // 2) Per-(node,head) rank-1 projections + init of max/denom/out.
    {
        const int nt = nNodes * HEADS;
        node_prep_kernel<<<(nt + 255) / 256, 256, 0, stream>>>(
            hw, Wa, asrc, adst, mmax, denom, out, nNodes);
    }

    // 3) Edge logits + segment max.
    edge_alpha_kernel<<<(nE + 255) / 256, 256, 0, stream>>>(
        src, dst, asrc, adst, ba, alpha, mmax, nE);

    // 4) exp(alpha - max) + segment sum.
    edge_exp_kernel<<<(nE + 255) / 256, 256, 0, stream>>>(
        dst, alpha, mmax, denom, nE);

    // 5) Normalized weighted scatter of src features to dst.
    {
        const int nt = nE * OUT_DIM;   // 51.2M
        edge_scatter_kernel<<<(nt + 255) / 256, 256, 0, stream>>>(
            src, dst, alpha, denom, hw, out, nE);
    }
}